// SimpleNodeEmbedder_63041529971089
// MI455X (gfx1250) — compile-verified
//
#include <hip/hip_runtime.h>

// ---------------------------------------------------------------------------
// Problem constants (match the JAX reference)
// ---------------------------------------------------------------------------
#define K_B    16
#define K_N    2048
#define K_S    4096
#define K_D    128
#define K_NNZ  (1 << 20)
#define OUT_FLOATS (K_B * K_N * K_D)   // 4,194,304 floats = 16 MB

#define PIPE   4                        // async pipeline depth (slots per wave)

// ---------------------------------------------------------------------------
// Helpers
// ---------------------------------------------------------------------------
__device__ __forceinline__ unsigned lds_byte_addr(const void* p) {
  // generic -> LDS(AS3) addrspacecast, then ptrtoint: byte offset from the
  // wave's LDS base.
  return (unsigned)(unsigned long long)
      (const __attribute__((address_space(3))) char*)p;
}

// One per-lane 16-B async copy global -> LDS (wave total: 512 B row).
__device__ __forceinline__ void async_load_row16(unsigned lds_off,
                                                 const float* gsrc) {
  unsigned long long ga = (unsigned long long)gsrc;
  asm volatile("global_load_async_to_lds_b128 %0, %1, off"
               :
               : "v"(lds_off), "v"(ga)
               : "memory");
}

// One per-lane 16-B async copy LDS -> global.
__device__ __forceinline__ void async_store_row16(float* gdst,
                                                  unsigned lds_off) {
  unsigned long long ga = (unsigned long long)gdst;
  asm volatile("global_store_async_from_lds_b128 %0, %1, off"
               :
               : "v"(ga), "v"(lds_off)
               : "memory");
}

__device__ __forceinline__ void wait_async_le3() {
  asm volatile("s_wait_asynccnt 0x3" ::: "memory");
}
__device__ __forceinline__ void wait_async_0() {
  asm volatile("s_wait_asynccnt 0x0" ::: "memory");
}

// ---------------------------------------------------------------------------
// Kernel 1: zero-fill the 16 MB output through the async-store engine.
// ---------------------------------------------------------------------------
__global__ __launch_bounds__(256) void nodepool_zero_async(float* __restrict__ out,
                                                           int n4) {
  __shared__ __align__(16) float zbuf[256 * 4];   // 4 KB of zeros

  ((float4*)zbuf)[threadIdx.x] = make_float4(0.f, 0.f, 0.f, 0.f);
  __syncthreads();  // LDS zeros visible before the async engine reads them

  const unsigned lds = lds_byte_addr(&zbuf[threadIdx.x * 4]);

  int i = blockIdx.x * blockDim.x + threadIdx.x;
  const int stride = gridDim.x * blockDim.x;
  for (; i < n4; i += stride) {
    async_store_row16(out + (size_t)i * 4, lds);
  }
  wait_async_0();   // drain ASYNCcnt (S_ENDPGM would also wait-idle)
}

// ---------------------------------------------------------------------------
// Kernel 2: fused gather + scale + scatter-add, 4-deep async-to-LDS pipeline.
//
// One wave32 per nnz entry: lane L handles floats [4L, 4L+4) of the 128-wide
// embedding row. Wave-uniform index chain -> scalar loads. Rows stream into a
// 4-slot LDS ring via global_load_async_to_lds_b128; s_wait_asynccnt 3
// releases the oldest row (async loads complete in order), it is consumed
// with ds_load_b128 and scattered with 4 global_atomic_add_f32 per lane into
// the L2-resident (16 MB << 192 MB) output. The reissue into a slot follows
// the atomics that consumed it (memory ops cannot cross the asm memory
// clobber), so the ds_load has completed (dscnt wait before use) before the
// async engine can overwrite the slot.
// ---------------------------------------------------------------------------
__global__ __launch_bounds__(256) void nodepool_scatter(
    const int*   __restrict__ subnode_ids,   // [B*S]
    const int*   __restrict__ mask_batch,    // [NNZ]
    const int*   __restrict__ mask_node,     // [NNZ]
    const int*   __restrict__ mask_subnode,  // [NNZ]
    const float* __restrict__ mask_values,   // [NNZ]
    const float* __restrict__ emb_table,     // [VOCAB*D]
    float*       __restrict__ out)           // [B*N*D]
{
  __shared__ __align__(16) float stage[8][PIPE][K_D];  // 8 waves x 4 slots x 512 B

  const int lane = (int)(threadIdx.x & 31u);
  const int wave = __builtin_amdgcn_readfirstlane((int)(threadIdx.x >> 5));
  const int wgid = (int)blockIdx.x * 8 + wave;      // blockDim fixed at 256 = 8 waves
  const int nw   = (int)gridDim.x * 8;

  const int epw = (K_NNZ + nw - 1) / nw;            // entries per wave (contiguous)
  const int e0  = wgid * epw;
  int e1 = e0 + epw;
  if (e1 > K_NNZ) e1 = K_NNZ;
  if (e0 >= e1) return;

  // Warm the index streams for this wave's chunk (global_prefetch_b8).
  __builtin_prefetch(mask_batch   + e0, 0, 0);
  __builtin_prefetch(mask_node    + e0, 0, 0);
  __builtin_prefetch(mask_subnode + e0, 0, 0);
  __builtin_prefetch(mask_values  + e0, 0, 0);

  // Generic fallback for tiny ranges (never hit with these launch params).
  if (e1 - e0 < PIPE) {
    for (int e = e0; e < e1; ++e) {
      const int b   = __builtin_amdgcn_readfirstlane(mask_batch[e]);
      const int nd  = __builtin_amdgcn_readfirstlane(mask_node[e]);
      const int sn  = __builtin_amdgcn_readfirstlane(mask_subnode[e]);
      const int tok = __builtin_amdgcn_readfirstlane(subnode_ids[b * K_S + sn]);
      const float v = mask_values[e];
      const float4 r = *((const float4*)(emb_table + tok * K_D) + lane);
      float* dst = out + (b * K_N + nd) * K_D + (lane << 2);
      __hip_atomic_fetch_add(dst + 0, r.x * v, __ATOMIC_RELAXED, __HIP_MEMORY_SCOPE_AGENT);
      __hip_atomic_fetch_add(dst + 1, r.y * v, __ATOMIC_RELAXED, __HIP_MEMORY_SCOPE_AGENT);
      __hip_atomic_fetch_add(dst + 2, r.z * v, __ATOMIC_RELAXED, __HIP_MEMORY_SCOPE_AGENT);
      __hip_atomic_fetch_add(dst + 3, r.w * v, __ATOMIC_RELAXED, __HIP_MEMORY_SCOPE_AGENT);
    }
    return;
  }

  // Per-lane LDS byte address of this wave's slot 0; slot stride = 512 B.
  const unsigned lds_base = lds_byte_addr(&stage[wave][0][lane * 4]);

  // Phase-indexed register queue carrying (batch, node) from issue to consume.
  int qb[PIPE], qn[PIPE];

  // ---- prologue: fill all 4 pipeline slots -------------------------------
#pragma unroll
  for (int p = 0; p < PIPE; ++p) {
    const int e   = e0 + p;
    const int b   = __builtin_amdgcn_readfirstlane(mask_batch[e]);
    const int nd  = __builtin_amdgcn_readfirstlane(mask_node[e]);
    const int sn  = __builtin_amdgcn_readfirstlane(mask_subnode[e]);
    const int tok = __builtin_amdgcn_readfirstlane(subnode_ids[b * K_S + sn]);
    qb[p] = b;
    qn[p] = nd;
    async_load_row16(lds_base + (unsigned)p * (K_D * 4), emb_table + tok * K_D + lane * 4);
  }

  // ---- steady state: consume e, reissue e+PIPE into the freed slot -------
  int e = e0;
#pragma unroll 4
  for (; e + PIPE < e1; ++e) {
    const int ph = e & (PIPE - 1);        // e0 is a multiple of PIPE

    wait_async_le3();                     // oldest async load (entry e) landed

    const float v  = mask_values[e];
    const float4 r = *(const float4*)&stage[wave][ph][lane * 4];
    float* dst = out + (qb[ph] * K_N + qn[ph]) * K_D + (lane << 2);
    __hip_atomic_fetch_add(dst + 0, r.x * v, __ATOMIC_RELAXED, __HIP_MEMORY_SCOPE_AGENT);
    __hip_atomic_fetch_add(dst + 1, r.y * v, __ATOMIC_RELAXED, __HIP_MEMORY_SCOPE_AGENT);
    __hip_atomic_fetch_add(dst + 2, r.z * v, __ATOMIC_RELAXED, __HIP_MEMORY_SCOPE_AGENT);
    __hip_atomic_fetch_add(dst + 3, r.w * v, __ATOMIC_RELAXED, __HIP_MEMORY_SCOPE_AGENT);

    // Reissue into the slot just consumed (entry e + PIPE).
    const int en  = e + PIPE;
    const int b   = __builtin_amdgcn_readfirstlane(mask_batch[en]);
    const int nd  = __builtin_amdgcn_readfirstlane(mask_node[en]);
    const int sn  = __builtin_amdgcn_readfirstlane(mask_subnode[en]);
    const int tok = __builtin_amdgcn_readfirstlane(subnode_ids[b * K_S + sn]);
    qb[ph] = b;
    qn[ph] = nd;
    async_load_row16(lds_base + (unsigned)ph * (K_D * 4), emb_table + tok * K_D + lane * 4);
  }

  // ---- tail: drain once, consume the last PIPE slots wait-free -----------
  wait_async_0();
#pragma unroll
  for (int p = 0; p < PIPE; ++p, ++e) {
    const int ph = e & (PIPE - 1);
    const float v  = mask_values[e];
    const float4 r = *(const float4*)&stage[wave][ph][lane * 4];
    float* dst = out + (qb[ph] * K_N + qn[ph]) * K_D + (lane << 2);
    __hip_atomic_fetch_add(dst + 0, r.x * v, __ATOMIC_RELAXED, __HIP_MEMORY_SCOPE_AGENT);
    __hip_atomic_fetch_add(dst + 1, r.y * v, __ATOMIC_RELAXED, __HIP_MEMORY_SCOPE_AGENT);
    __hip_atomic_fetch_add(dst + 2, r.z * v, __ATOMIC_RELAXED, __HIP_MEMORY_SCOPE_AGENT);
    __hip_atomic_fetch_add(dst + 3, r.w * v, __ATOMIC_RELAXED, __HIP_MEMORY_SCOPE_AGENT);
  }
}

// ---------------------------------------------------------------------------
// Host entry
// ---------------------------------------------------------------------------
extern "C" void kernel_launch(void* const* d_in, const int* in_sizes, int n_in,
                              void* d_out, int out_size, void* d_ws, size_t ws_size,
                              hipStream_t stream) {
  (void)in_sizes; (void)n_in; (void)out_size; (void)d_ws; (void)ws_size;

  const int*   subnode_ids  = (const int*)  d_in[0];
  const int*   mask_batch   = (const int*)  d_in[1];
  const int*   mask_node    = (const int*)  d_in[2];
  const int*   mask_subnode = (const int*)  d_in[3];
  const float* mask_values  = (const float*)d_in[4];
  const float* emb_table    = (const float*)d_in[5];
  float*       out          = (float*)      d_out;

  // 1) Zero the 16 MB output via the async-store engine (stays L2-resident).
  nodepool_zero_async<<<2048, 256, 0, stream>>>(out, OUT_FLOATS / 4);

  // 2) 2048 blocks * 8 waves = 16384 waves, 64 contiguous entries each.
  nodepool_scatter<<<2048, 256, 0, stream>>>(
      subnode_ids, mask_batch, mask_node, mask_subnode, mask_values, emb_table, out);
}